// RelationalGraphConvolution_38826504356516
// MI455X (gfx1250) — compile-verified
//
#include <hip/hip_runtime.h>
#include <hip/hip_bf16.h>

typedef __attribute__((ext_vector_type(2))) float v2f;
typedef __attribute__((ext_vector_type(4))) float v4f;
typedef __attribute__((ext_vector_type(8))) float v8f;

#define N_NODES 50000
#define N_REL   8
#define N_EDGES 60000
#define F_IN    128
#define UNITS   128

// LDS layout for W: pair-rows (k/2), each holding (W[k][c], W[k+1][c]) adjacent.
// Stride 256+32 floats so the two half-wave k-rows hit disjoint bank halves.
#define WPAIR_STRIDE 288
// LDS layout for the staged A strip: row stride 132 floats (= 0 mod 4 so b128
// stores stay 16B-aligned; bank index = (4*row + k) mod 64 so the two
// half-wave k-columns of a fragment load hit disjoint bank sets).
#define A_STRIDE 132

// ---------------------------------------------------------------------------
// GEMM: out[row, :] (+)= A[row, :] @ W, A:[nrows,128], W:[128,128], fp32 WMMA.
// Block = 128 threads = 4 waves; each wave computes a 16x128 strip.
// A strip and W are staged in LDS with coalesced b128 global loads.
// ---------------------------------------------------------------------------
__global__ __launch_bounds__(128) void rgcn_gemm_kernel(
    const float* __restrict__ A,
    const float* __restrict__ W,
    float* __restrict__ out,
    int accum, int nrows)
{
    __shared__ float Wl[64 * WPAIR_STRIDE];   // 73,728 B
    __shared__ float Al[64 * A_STRIDE];       // 33,792 B

    const int t = threadIdx.x;
    const int row_base = blockIdx.x * 64;

    // Stage W into paired layout; global side coalesced b128.
#pragma unroll
    for (int p = 0; p < 32; ++p) {
        int i  = p * 128 + t;          // float4 index in [0, 4096)
        int k  = i >> 5;               // W row
        int c4 = (i & 31) * 4;         // W col
        v4f wv = *(const v4f*)(W + k * UNITS + c4);
        float* dstp = &Wl[(k >> 1) * WPAIR_STRIDE + c4 * 2 + (k & 1)];
        dstp[0] = wv.x; dstp[2] = wv.y; dstp[4] = wv.z; dstp[6] = wv.w;
    }

    // Stage this block's 64-row A strip; coalesced b128, tail-guarded.
#pragma unroll
    for (int p = 0; p < 16; ++p) {
        int i  = p * 128 + t;          // float4 index in [0, 2048)
        int r  = i >> 5;
        int c4 = (i & 31) * 4;
        if (row_base + r < nrows)
            *(v4f*)&Al[r * A_STRIDE + c4] =
                *(const v4f*)(A + (size_t)(row_base + r) * F_IN + c4);
    }
    __syncthreads();

    const int wave = t >> 5;
    const int lane = t & 31;
    const int row0 = row_base + wave * 16;
    if (row0 >= nrows) return;   // wave-uniform: EXEC stays all-ones below

    const int half = lane >> 4;      // 0: lanes 0-15, 1: lanes 16-31
    const int n0   = lane & 15;
    const int koff = half * 2;       // this half's k sub-rows for A and B
    const float* aptr = &Al[(wave * 16 + n0) * A_STRIDE + koff];

    v8f acc[8];
    if (accum) {
#pragma unroll
        for (int nt = 0; nt < 8; ++nt)
#pragma unroll
            for (int v = 0; v < 8; ++v)
                acc[nt][v] = out[(size_t)(row0 + v + half * 8) * UNITS + nt * 16 + n0];
    } else {
#pragma unroll
        for (int nt = 0; nt < 8; ++nt)
            acc[nt] = (v8f)0.0f;
    }

#pragma unroll 4
    for (int k = 0; k < F_IN; k += 4) {
        // A 16x4 tile: lane<16 -> (k,k+1) of its row; lane>=16 -> (k+2,k+3)
        v2f a = *(const v2f*)(aptr + k);
        const float* bbase = &Wl[((k + koff) >> 1) * WPAIR_STRIDE + n0 * 2];
#pragma unroll
        for (int nt = 0; nt < 8; ++nt) {
            // B 4x16 tile for output cols [nt*16, nt*16+16)
            v2f b = *(const v2f*)(bbase + nt * 32);
            acc[nt] = __builtin_amdgcn_wmma_f32_16x16x4_f32(
                false, a, false, b, (short)0, acc[nt], false, false);
        }
    }

#pragma unroll
    for (int nt = 0; nt < 8; ++nt)
#pragma unroll
        for (int v = 0; v < 8; ++v)
            out[(size_t)(row0 + v + half * 8) * UNITS + nt * 16 + n0] = acc[nt][v];
}

// ---------------------------------------------------------------------------
// Zero the [N,128] message buffer.
// ---------------------------------------------------------------------------
__global__ __launch_bounds__(256) void rgcn_zero_kernel(v4f* __restrict__ p, int n4)
{
    int i = blockIdx.x * 256 + threadIdx.x;
    if (i < n4) p[i] = (v4f)0.0f;
}

// ---------------------------------------------------------------------------
// Scatter: m[dst_e, :] += w_e * x[src_e, :]. One wave per edge, 4 floats/lane.
// ---------------------------------------------------------------------------
__global__ __launch_bounds__(128) void rgcn_scatter_kernel(
    const float* __restrict__ x,
    const int* __restrict__ src,
    const int* __restrict__ dst,
    const float* __restrict__ w,
    float* __restrict__ m, int nedges)
{
    const int wave = threadIdx.x >> 5;
    const int lane = threadIdx.x & 31;
    const int e = blockIdx.x * 4 + wave;
    if (e >= nedges) return;

    const int s = src[e];
    const int d = dst[e];
    const float we = w[e];

    v4f v = *(const v4f*)(x + (size_t)s * F_IN + lane * 4);
    float* mp = m + (size_t)d * F_IN + lane * 4;
    atomicAdd(mp + 0, we * v.x);
    atomicAdd(mp + 1, we * v.y);
    atomicAdd(mp + 2, we * v.z);
    atomicAdd(mp + 3, we * v.w);
}

// ---------------------------------------------------------------------------
// Epilogue: out = relu(out + bias). One thread per 4 consecutive floats.
// ---------------------------------------------------------------------------
__global__ __launch_bounds__(256) void rgcn_bias_relu_kernel(
    float* __restrict__ out, const float* __restrict__ bias)
{
    int i = blockIdx.x * 256 + threadIdx.x;        // i in [0, N*32)
    if (i >= N_NODES * (UNITS / 4)) return;
    int c4 = (i & (UNITS / 4 - 1)) * 4;
    v4f v = *(v4f*)(out + (size_t)i * 4);
    v.x = fmaxf(v.x + bias[c4 + 0], 0.0f);
    v.y = fmaxf(v.y + bias[c4 + 1], 0.0f);
    v.z = fmaxf(v.z + bias[c4 + 2], 0.0f);
    v.w = fmaxf(v.w + bias[c4 + 3], 0.0f);
    *(v4f*)(out + (size_t)i * 4) = v;
}

extern "C" void kernel_launch(void* const* d_in, const int* in_sizes, int n_in,
                              void* d_out, int out_size, void* d_ws, size_t ws_size,
                              hipStream_t stream) {
    const float* features    = (const float*)d_in[0];  // [1, N, 128]
    const int*   edge_src    = (const int*)  d_in[1];  // [8, E]
    const int*   edge_dst    = (const int*)  d_in[2];  // [8, E]
    const float* edge_w      = (const float*)d_in[3];  // [8, E]
    const float* self_kernel = (const float*)d_in[4];  // [128, 128]
    const float* rel_kernels = (const float*)d_in[5];  // [8, 128, 128]
    const float* bias        = (const float*)d_in[6];  // [128]
    float* out = (float*)d_out;                        // [1, N, 128]
    float* m   = (float*)d_ws;                         // [N, 128] scratch (25.6 MB)

    const dim3 gemm_grid((N_NODES + 63) / 64);
    const int n4 = N_NODES * F_IN / 4;
    const dim3 zero_grid((n4 + 255) / 256);
    const dim3 scat_grid((N_EDGES + 3) / 4);

    // Self-loop term: out = x @ self_kernel
    rgcn_gemm_kernel<<<gemm_grid, 128, 0, stream>>>(features, self_kernel, out,
                                                    /*accum=*/0, N_NODES);

    // Per relation: m = segsum_dst(w_e * x[src_e]); out += m @ W_r
    for (int r = 0; r < N_REL; ++r) {
        rgcn_zero_kernel<<<zero_grid, 256, 0, stream>>>((v4f*)m, n4);
        rgcn_scatter_kernel<<<scat_grid, 128, 0, stream>>>(
            features, edge_src + r * N_EDGES, edge_dst + r * N_EDGES,
            edge_w + r * N_EDGES, m, N_EDGES);
        rgcn_gemm_kernel<<<gemm_grid, 128, 0, stream>>>(
            m, rel_kernels + (size_t)r * F_IN * UNITS, out, /*accum=*/1, N_NODES);
    }

    // out = relu(out + bias)
    rgcn_bias_relu_kernel<<<dim3((N_NODES * (UNITS / 4) + 255) / 256), 256, 0, stream>>>(
        out, bias);
}